// SPGConvLayer_56684978372726
// MI455X (gfx1250) — compile-verified
//
#include <hip/hip_runtime.h>

#define N_NODES 100000
#define N_EDGES 1600000
#define BN_EPS  1e-5f

typedef __attribute__((ext_vector_type(2))) float v2f;
typedef __attribute__((ext_vector_type(8))) float v8f;

// workspace layout (float offsets)
#define WS_H0    0
#define WS_H1    (N_NODES * 64)
#define WS_M     (2 * N_NODES * 64)       // M0[4096] M1[4096] (folded weights)
#define WS_SUM   (WS_M + 8192)            // 64
#define WS_SQ    (WS_SUM + 64)            // 64
#define WS_SCALE (WS_SQ + 64)             // 64
#define WS_SHIFT (WS_SCALE + 64)          // 64

__global__ void zero_ws(float* ws, long n) {
    long i = (long)blockIdx.x * blockDim.x + threadIdx.x;
    long stride = (long)gridDim.x * blockDim.x;
    for (; i < n; i += stride) ws[i] = 0.0f;
}

// Fold MLP into edge weights: M_k[i][c] = sum_o linear[k][i][o] * mlp_w[c][o]
__global__ void prep_weights(const float* __restrict__ linear,
                             const float* __restrict__ mlp_w,
                             float* __restrict__ Mw) {
    for (int idx = threadIdx.x; idx < 8192; idx += blockDim.x) {
        int k = idx >> 12, rem = idx & 4095, i = rem >> 6, c = rem & 63;
        const float* wrow = linear + k * 4096 + i * 64;
        const float* mrow = mlp_w + c * 64;
        float s = 0.0f;
#pragma unroll
        for (int o = 0; o < 64; ++o) s += wrow[o] * mrow[o];
        Mw[idx] = s;
    }
}

// One wave per edge: 32 lanes x float2 = 64 feats. Gather feature row, atomic
// scatter into per-order accumulator h0/h1. Everything L2-resident.
__global__ void __launch_bounds__(256) edge_scatter(
    const float* __restrict__ feat, const int* __restrict__ esrc,
    const int* __restrict__ edst, const int* __restrict__ eord,
    float* __restrict__ h0, float* __restrict__ h1) {
    int wave = (int)((blockIdx.x * blockDim.x + threadIdx.x) >> 5);
    int lane = threadIdx.x & 31;
    if (wave >= N_EDGES) return;
    int src = esrc[wave];
    int dst = edst[wave];
    int ord = eord[wave];
    const float2 v = *(const float2*)(feat + (long)src * 64 + lane * 2);
    float* h = ord ? h1 : h0;
    float* p = h + (long)dst * 64 + lane * 2;
    atomicAdd(p, v.x);
    atomicAdd(p + 1, v.y);
}

// y = relu([h0|h1] @ [M0;M1] + bias), fp32 WMMA 16x16x4.
// Block = 8 waves covering a 32-node x 64-col tile; also accumulates
// per-column sum / sum-of-squares for BatchNorm.
__global__ void __launch_bounds__(256) gemm_bn_pass(
    const float* __restrict__ h0, const float* __restrict__ h1,
    const float* __restrict__ Mw, const float* __restrict__ bias,
    float* __restrict__ y, float* __restrict__ gsum, float* __restrict__ gsq) {
    __shared__ float sM[8192];   // M0 then M1, row-major [k][c]
    __shared__ float sB[64];
    __shared__ float lsum[64];
    __shared__ float lsq[64];

    int tid = threadIdx.x;
    for (int i = tid; i < 8192; i += 256) sM[i] = Mw[i];
    if (tid < 64) { sB[tid] = bias[tid]; lsum[tid] = 0.0f; lsq[tid] = 0.0f; }
    __syncthreads();

    int wave = tid >> 5, lane = tid & 31;
    int rowg = wave >> 2, colg = wave & 3;
    int node_base = blockIdx.x * 32 + rowg * 16;
    int col_base  = colg * 16;
    int m = lane & 15;          // A row within tile / D column index
    int khalf = lane >> 4;      // selects K-pair 0..1 or 2..3

    const float* a0 = h0 + (long)(node_base + m) * 64 + khalf * 2;
    const float* a1 = h1 + (long)(node_base + m) * 64 + khalf * 2;
    const float* b0 = sM +        khalf * 2 * 64 + col_base + m;
    const float* b1 = sM + 4096 + khalf * 2 * 64 + col_base + m;

    v8f acc = {0.f, 0.f, 0.f, 0.f, 0.f, 0.f, 0.f, 0.f};
#pragma unroll
    for (int ks = 0; ks < 16; ++ks) {   // K = 64 over h0/M0
        v2f a = *(const v2f*)(a0 + ks * 4);                 // K contiguous pair
        v2f b = { b0[ks * 4 * 64], b0[ks * 4 * 64 + 64] };  // rows k, k+1
        acc = __builtin_amdgcn_wmma_f32_16x16x4_f32(
            false, a, false, b, (short)0, acc, false, false);
    }
#pragma unroll
    for (int ks = 0; ks < 16; ++ks) {   // K = 64 over h1/M1
        v2f a = *(const v2f*)(a1 + ks * 4);
        v2f b = { b1[ks * 4 * 64], b1[ks * 4 * 64 + 64] };
        acc = __builtin_amdgcn_wmma_f32_16x16x4_f32(
            false, a, false, b, (short)0, acc, false, false);
    }

    // D layout: VGPR r -> M = khalf*8 + r, N = lane&15
    float bb = sB[col_base + m];
    float s1 = 0.0f, s2 = 0.0f;
#pragma unroll
    for (int r = 0; r < 8; ++r) {
        int mm = khalf * 8 + r;
        float v = acc[r] + bb;
        v = v > 0.0f ? v : 0.0f;
        y[(long)(node_base + mm) * 64 + col_base + m] = v;
        s1 += v;
        s2 += v * v;
    }
    atomicAdd(&lsum[col_base + m], s1);   // ds_add_f32
    atomicAdd(&lsq[col_base + m], s2);
    __syncthreads();
    if (tid < 64)       atomicAdd(&gsum[tid], lsum[tid]);
    else if (tid < 128) atomicAdd(&gsq[tid - 64], lsq[tid - 64]);
}

__global__ void bn_stats(const float* __restrict__ gsum, const float* __restrict__ gsq,
                         const float* __restrict__ gamma, const float* __restrict__ beta,
                         float* __restrict__ scale, float* __restrict__ shift) {
    int c = threadIdx.x;
    if (c < 64) {
        float mean = gsum[c] * (1.0f / N_NODES);
        float var  = gsq[c] * (1.0f / N_NODES) - mean * mean;
        float inv  = rsqrtf(var + BN_EPS);
        float sc   = gamma[c] * inv;
        scale[c] = sc;
        shift[c] = beta[c] - mean * sc;
    }
}

__global__ void bn_apply(float* __restrict__ y, const float* __restrict__ scale,
                         const float* __restrict__ shift) {
    long i = (long)blockIdx.x * blockDim.x + threadIdx.x;
    if (i < (long)N_NODES * 64) {
        int c = (int)(i & 63);
        y[i] = y[i] * scale[c] + shift[c];
    }
}

extern "C" void kernel_launch(void* const* d_in, const int* in_sizes, int n_in,
                              void* d_out, int out_size, void* d_ws, size_t ws_size,
                              hipStream_t stream) {
    const float* feature = (const float*)d_in[0];
    // d_in[1] sp_embeddings: unused by forward
    const float* linear  = (const float*)d_in[2];
    const float* mlp_w   = (const float*)d_in[3];
    const float* mlp_b   = (const float*)d_in[4];
    const float* gamma   = (const float*)d_in[5];
    const float* beta    = (const float*)d_in[6];
    const int*   esrc    = (const int*)d_in[7];
    const int*   edst    = (const int*)d_in[8];
    const int*   eord    = (const int*)d_in[9];

    float* ws    = (float*)d_ws;
    float* h0    = ws + WS_H0;
    float* h1    = ws + WS_H1;
    float* Mw    = ws + WS_M;
    float* gsum  = ws + WS_SUM;
    float* gsq   = ws + WS_SQ;
    float* scale = ws + WS_SCALE;
    float* shift = ws + WS_SHIFT;
    float* y     = (float*)d_out;

    long nzero = 2L * N_NODES * 64 + 8192 + 256;
    zero_ws<<<4096, 256, 0, stream>>>(ws, nzero);
    prep_weights<<<1, 256, 0, stream>>>(linear, mlp_w, Mw);
    edge_scatter<<<N_EDGES / 8, 256, 0, stream>>>(feature, esrc, edst, eord, h0, h1);
    gemm_bn_pass<<<N_NODES / 32, 256, 0, stream>>>(h0, h1, Mw, mlp_b, y, gsum, gsq);
    bn_stats<<<1, 64, 0, stream>>>(gsum, gsq, gamma, beta, scale, shift);
    bn_apply<<<(N_NODES * 64 + 255) / 256, 256, 0, stream>>>(y, scale, shift);
}